// GraphSAGE_Mean_10368051052691
// MI455X (gfx1250) — compile-verified
//
#include <hip/hip_runtime.h>
#include <hip/hip_bf16.h>
#include <stdint.h>

typedef float v2f __attribute__((ext_vector_type(2)));
typedef float v8f __attribute__((ext_vector_type(8)));

#define N_NODES 50000
#define N_EDGES 640000
#define C_IN    128
#define C_OUT   47

// ---------------------------------------------------------------------------
// 1) in-degree counts (mean denominator), f32 so the divide is cheap later
// ---------------------------------------------------------------------------
__global__ __launch_bounds__(256)
void count_kernel(const long long* __restrict__ ei, float* __restrict__ cnt) {
    int e = blockIdx.x * blockDim.x + threadIdx.x;
    if (e < N_EDGES) {
        int dst = (int)ei[N_EDGES + e];
        atomicAdd(&cnt[dst], 1.0f);
    }
}

// ---------------------------------------------------------------------------
// 2) scatter-sum: 32 lanes per edge, each lane moves 4 contiguous channels.
//    float4 gather from feat[src] (L2-resident), 4 f32 atomics into agg[dst].
// ---------------------------------------------------------------------------
__global__ __launch_bounds__(256)
void scatter_kernel(const float* __restrict__ feat,
                    const long long* __restrict__ ei,
                    float* __restrict__ agg) {
    long long tid = (long long)blockIdx.x * blockDim.x + threadIdx.x;
    int e  = (int)(tid >> 5);
    int c4 = (int)(tid & 31) << 2;
    if (e < N_EDGES) {
        int src = (int)ei[e];
        int dst = (int)ei[N_EDGES + e];
        const float4 v = *(const float4*)(feat + (size_t)src * C_IN + c4);
        float* p = agg + (size_t)dst * C_IN + c4;
        atomicAdd(p + 0, v.x);
        atomicAdd(p + 1, v.y);
        atomicAdd(p + 2, v.z);
        atomicAdd(p + 3, v.w);
    }
}

// ---------------------------------------------------------------------------
// 3) fused SAGE layer:  out = act( (agg/max(cnt,1)) @ Wl^T + self @ Wr^T + b )
//    One wave per 16-row stripe. A tiles AND W tiles staged in LDS (zero pad
//    resolved at staging), so the K-loop is pure ds_load_b64 + WMMA with
//    EXEC untouched. Exact fp32 via V_WMMA_F32_16X16X4_F32.
//
//    A layout (16x4 f32): lanes 0-15 -> M=l, K=k,k+1 ; lanes 16-31 -> K=k+2,k+3
//    B layout (4x16 f32): lanes 0-15 -> N=l, K=k,k+1 ; lanes 16-31 -> K=k+2,k+3
//    C/D (16x16 f32): vgpr j -> row M=j (lanes 0-15) / M=j+8 (lanes 16-31)
// ---------------------------------------------------------------------------
template<int NCOLT, bool RELU>
__global__ __launch_bounds__(32)
void sage_gemm_kernel(const float* __restrict__ agg,
                      const float* __restrict__ cnt,
                      const float* __restrict__ self,
                      const float* __restrict__ Wl,   // [nValid, 128] row-major
                      const float* __restrict__ Wr,   // [nValid, 128] row-major
                      const float* __restrict__ bias, // [nValid]
                      float* __restrict__ out,
                      int nValid, int ldOut) {
    __shared__ float sA [16 * C_IN];  // scaled aggregate tile
    __shared__ float sS [16 * C_IN];  // self-feature tile
    __shared__ float sWl[16 * C_IN];  // lin_l weight tile (zero-padded)
    __shared__ float sWr[16 * C_IN];  // lin_r weight tile (zero-padded)

    const int row0 = blockIdx.x * 16;
    const int lane = threadIdx.x;

    // cooperative, coalesced staging of A tiles: 2048 floats each / 32 lanes
    for (int idx = lane * 4; idx < 16 * C_IN; idx += 32 * 4) {
        int r = idx >> 7;
        int c = idx & (C_IN - 1);
        float invc = 1.0f / fmaxf(cnt[row0 + r], 1.0f);
        float4 va = *(const float4*)(agg  + (size_t)(row0 + r) * C_IN + c);
        float4 vs = *(const float4*)(self + (size_t)(row0 + r) * C_IN + c);
        va.x *= invc; va.y *= invc; va.z *= invc; va.w *= invc;
        *(float4*)(sA + idx) = va;
        *(float4*)(sS + idx) = vs;
    }

    const int half = lane >> 4;   // 0: K lo pair, 1: K hi pair
    const int l    = lane & 15;   // A row / B col within tile

    for (int ct = 0; ct < NCOLT; ++ct) {
        const int col0 = ct * 16;

        // prefetch next tile's weight rows while we work on this one
        if (ct + 1 < NCOLT) {
            int nrow = col0 + 16 + l;
            if (nrow < nValid) {
                __builtin_prefetch(Wl + (size_t)nrow * C_IN, 0, 1);
                __builtin_prefetch(Wr + (size_t)nrow * C_IN, 0, 1);
            }
        }

        // stage weight tiles, zero-padding rows >= nValid (resolved here,
        // NOT in the K-loop -> no EXEC games around the WMMAs)
        for (int idx = lane * 4; idx < 16 * C_IN; idx += 32 * 4) {
            int r = idx >> 7;
            int c = idx & (C_IN - 1);
            int wrow = col0 + r;
            float4 zl = {0.f, 0.f, 0.f, 0.f};
            float4 zr = {0.f, 0.f, 0.f, 0.f};
            if (wrow < nValid) {
                zl = *(const float4*)(Wl + (size_t)wrow * C_IN + c);
                zr = *(const float4*)(Wr + (size_t)wrow * C_IN + c);
            }
            *(float4*)(sWl + idx) = zl;
            *(float4*)(sWr + idx) = zr;
        }
        __syncthreads();

        const float* pa = sA  + l * C_IN;
        const float* ps = sS  + l * C_IN;
        const float* pl = sWl + l * C_IN;
        const float* pr = sWr + l * C_IN;

        v8f acc = {0.f, 0.f, 0.f, 0.f, 0.f, 0.f, 0.f, 0.f};

        #pragma unroll 8
        for (int k = 0; k < C_IN; k += 4) {
            const int ka = k + half * 2;            // 8-byte aligned
            v2f a1 = *(const v2f*)(pa + ka);
            v2f b1 = *(const v2f*)(pl + ka);
            acc = __builtin_amdgcn_wmma_f32_16x16x4_f32(
                      false, a1, false, b1, (short)0, acc, false, false);
            v2f a2 = *(const v2f*)(ps + ka);
            v2f b2 = *(const v2f*)(pr + ka);
            acc = __builtin_amdgcn_wmma_f32_16x16x4_f32(
                      false, a2, false, b2, (short)0, acc, false, false);
        }

        const int ocol = col0 + l;
        if (ocol < nValid) {
            const float bv = bias[ocol];
            #pragma unroll
            for (int j = 0; j < 8; ++j) {
                int orow = row0 + j + half * 8;
                float v = acc[j] + bv;
                if (RELU) v = fmaxf(v, 0.0f);
                out[(size_t)orow * ldOut + ocol] = v;
            }
        }
        __syncthreads();   // protect W tiles before next iteration overwrites
    }
}

// ---------------------------------------------------------------------------
// 4) row-wise log_softmax, one wave32 per row (47 cols -> 2 elems/lane max)
// ---------------------------------------------------------------------------
__global__ __launch_bounds__(256)
void logsoftmax_kernel(float* __restrict__ out) {
    int wid  = threadIdx.x >> 5;
    int lane = threadIdx.x & 31;
    int row  = blockIdx.x * 8 + wid;
    if (row >= N_NODES) return;

    float* p = out + (size_t)row * C_OUT;
    float v0 = (lane      < C_OUT) ? p[lane]      : -__builtin_inff();
    float v1 = (lane + 32 < C_OUT) ? p[lane + 32] : -__builtin_inff();

    float m = fmaxf(v0, v1);
    #pragma unroll
    for (int off = 16; off > 0; off >>= 1)
        m = fmaxf(m, __shfl_xor(m, off, 32));

    float s = ((lane      < C_OUT) ? expf(v0 - m) : 0.0f)
            + ((lane + 32 < C_OUT) ? expf(v1 - m) : 0.0f);
    #pragma unroll
    for (int off = 16; off > 0; off >>= 1)
        s += __shfl_xor(s, off, 32);

    float lg = logf(s);
    if (lane      < C_OUT) p[lane]      = v0 - m - lg;
    if (lane + 32 < C_OUT) p[lane + 32] = v1 - m - lg;
}

// ---------------------------------------------------------------------------
extern "C" void kernel_launch(void* const* d_in, const int* in_sizes, int n_in,
                              void* d_out, int out_size, void* d_ws, size_t ws_size,
                              hipStream_t stream) {
    const float*     x   = (const float*)d_in[0];
    const long long* ei  = (const long long*)d_in[1];  // int64 [2, E]
    const float*     W1l = (const float*)d_in[2];
    const float*     W1r = (const float*)d_in[3];
    const float*     b1  = (const float*)d_in[4];
    const float*     W2l = (const float*)d_in[5];
    const float*     W2r = (const float*)d_in[6];
    const float*     b2  = (const float*)d_in[7];
    float* out = (float*)d_out;

    float* agg = (float*)d_ws;                          // 6,400,000 f32
    float* h   = agg + (size_t)N_NODES * C_IN;          // 6,400,000 f32
    float* cnt = h   + (size_t)N_NODES * C_IN;          //    50,000 f32

    const size_t aggBytes = (size_t)N_NODES * C_IN * sizeof(float);

    // ---- layer 1 ----
    hipMemsetAsync(agg, 0, aggBytes, stream);
    hipMemsetAsync(cnt, 0, (size_t)N_NODES * sizeof(float), stream);
    count_kernel<<<(N_EDGES + 255) / 256, 256, 0, stream>>>(ei, cnt);
    {
        long long t = (long long)N_EDGES * 32;
        scatter_kernel<<<(unsigned)((t + 255) / 256), 256, 0, stream>>>(x, ei, agg);
    }
    sage_gemm_kernel<8, true><<<N_NODES / 16, 32, 0, stream>>>(
        agg, cnt, x, W1l, W1r, b1, h, C_IN, C_IN);

    // ---- layer 2 ----
    hipMemsetAsync(agg, 0, aggBytes, stream);
    {
        long long t = (long long)N_EDGES * 32;
        scatter_kernel<<<(unsigned)((t + 255) / 256), 256, 0, stream>>>(h, ei, agg);
    }
    sage_gemm_kernel<3, false><<<N_NODES / 16, 32, 0, stream>>>(
        agg, cnt, h, W2l, W2r, b2, out, C_OUT, C_OUT);

    // ---- log_softmax ----
    logsoftmax_kernel<<<(N_NODES + 7) / 8, 256, 0, stream>>>(out);
}